// QSoftmax_22600117912050
// MI455X (gfx1250) — compile-verified
//
#include <hip/hip_runtime.h>
#include <stdint.h>

// QSoftmax (8-bit LUT softmax) for MI455X / gfx1250.
// B=8, S=2048 -> 16384 rows of 2048 int32 indices; 256-entry f32 LUT.
// Memory-bound: 128 MiB read + 128 MiB write => ~11 us floor @ 23.3 TB/s.
// Strategy: one wave32 per row; LUT staged to LDS via the Tensor Data Mover;
// row sum reduced across the wave using chained v_wmma_f32_16x16x4_f32 with a
// ones B-matrix (exact f32 accumulation, no shuffles).

typedef __attribute__((ext_vector_type(2))) float v2f;
typedef __attribute__((ext_vector_type(8))) float v8f;
typedef __attribute__((ext_vector_type(4))) unsigned int v4u;
typedef __attribute__((ext_vector_type(8))) int v8i;
typedef __attribute__((ext_vector_type(4))) int v4i;

#define SEQ 2048
#define NROWS (8 * 2048)
#define WAVES_PER_BLOCK 8
#define ITER 16 /* 16 x int4 per lane = 64 elements; 32 lanes * 64 = 2048 */

__global__ __launch_bounds__(256) void qsoftmax_gfx1250(
    const int* __restrict__ inp, const float* __restrict__ lut,
    const float* __restrict__ exp_scale_p, const float* __restrict__ exp_zero_p,
    const float* __restrict__ normed_scale_p,
    const float* __restrict__ normed_zero_p, float* __restrict__ out) {
  __shared__ float lutLDS[256];  // sole LDS object -> LDS offset 0

  const int tid = threadIdx.x;

  // ---- Stage the 1 KiB LUT into LDS with the TDM (async-tensor path) ----
#if defined(__gfx1250__) && __has_builtin(__builtin_amdgcn_tensor_load_to_lds) && \
    __has_builtin(__builtin_amdgcn_s_wait_tensorcnt)
  if (tid < 32) {  // wave 0 issues one TENSOR_LOAD_TO_LDS for the block
    const uint64_t ga = (uint64_t)(uintptr_t)lut;
    v4u g0;
    g0.x = 1u;                                        // count=1, is_restore=0
    g0.y = 0u;                                        // lds_addr = 0 (lutLDS)
    g0.z = (unsigned)(ga & 0xFFFFFFFFu);              // global_addr[31:0]
    g0.w = (unsigned)((ga >> 32) & 0x01FFFFFFu)       // global_addr[56:32]
           | (2u << 30);                              // type = 2 ("image")
    v8i g1;
    g1[0] = (int)(2u << 16);    // data_size=2 -> 4 bytes; wg_mask=0
    g1[1] = (int)(256u << 16);  // tensor_dim0[15:0] = 256 in bits[63:48]
    g1[2] = (int)(1u << 16);    // tensor_dim0 hi=0; tensor_dim1[15:0]=1
    g1[3] = (int)(256u << 16);  // tensor_dim1 hi=0; tile_dim0=256
    g1[4] = 1;                  // tile_dim1=1, tile_dim2=0
    g1[5] = 256;                // tensor_dim0_stride = 256 (data_size units)
    g1[6] = 0;
    g1[7] = 0;
    const v4i gz4 = {0, 0, 0, 0};                // groups 2/3 unused (<=2D)
    const v8i gz8 = {0, 0, 0, 0, 0, 0, 0, 0};    // trailing group (clang-23 form)
    __builtin_amdgcn_tensor_load_to_lds(g0, g1, gz4, gz4, gz8, 0);
    __builtin_amdgcn_s_wait_tensorcnt(0);
  }
#else
  lutLDS[tid & 255] = lut[tid & 255];
#endif
  __syncthreads();

  // The TDM wrote lutLDS behind the compiler's back (descriptor holds the LDS
  // address as an integer). Escape the address into an asm with a memory
  // clobber so LLVM cannot fold the (apparently unwritten) LDS loads to undef.
  asm volatile("" : : "r"(static_cast<const void*>(&lutLDS[0])) : "memory");

  const int wave = tid >> 5;
  const int lane = tid & 31;
  const int row = blockIdx.x * WAVES_PER_BLOCK + wave;
  const size_t base = (size_t)row * SEQ;
  const int4* __restrict__ inRow = (const int4*)(inp + base);

  // ---- Stream row: coalesced b128 loads, LDS LUT gathers, cache expd ----
  float e[4 * ITER];
  float p = 0.0f;
#pragma unroll
  for (int i = 0; i < ITER; ++i) {
    const int4 v = inRow[i * 32 + lane];
    const float e0 = lutLDS[v.x & 255];
    const float e1 = lutLDS[v.y & 255];
    const float e2 = lutLDS[v.z & 255];
    const float e3 = lutLDS[v.w & 255];
    e[4 * i + 0] = e0;
    e[4 * i + 1] = e1;
    e[4 * i + 2] = e2;
    e[4 * i + 3] = e3;
    p += (e0 + e1) + (e2 + e3);
  }

  // ---- Full wave32 reduction of partials via f32 WMMA (B = ones) ----
  // WMMA1: A[m][0]=p_m (lanes 0-15), A[m][2]=p_{m+16} (lanes 16-31), rest 0.
  //        D1[m][n] = p_m + p_{m+16}  (same for every column n).
  // WMMA2..5: feed D1 VGPR pairs back as A with accumulate:
  //        pair (2j,2j+1) contributes q_{2j}+q_{2j+1}+q_{2j+8}+q_{2j+9},
  //        summing all 16 q's -> total in every element of acc.
  float rowsum;
#if __has_builtin(__builtin_amdgcn_wmma_f32_16x16x4_f32)
  {
    v2f a;
    a.x = p;
    a.y = 0.0f;
    v2f ones;
    ones.x = 1.0f;
    ones.y = 1.0f;
    v8f zero = {0.f, 0.f, 0.f, 0.f, 0.f, 0.f, 0.f, 0.f};
    v8f d1 = __builtin_amdgcn_wmma_f32_16x16x4_f32(false, a, false, ones,
                                                   (short)0, zero, false, false);
    v8f acc = zero;
#pragma unroll
    for (int j = 0; j < 8; j += 2) {
      v2f a2;
      a2.x = d1[j];
      a2.y = d1[j + 1];
      acc = __builtin_amdgcn_wmma_f32_16x16x4_f32(false, a2, false, ones,
                                                  (short)0, acc, false, false);
    }
    rowsum = acc[0];
  }
#else
  rowsum = p;
  for (int m = 16; m >= 1; m >>= 1) rowsum += __shfl_xor(rowsum, m, 32);
#endif

  // ---- Quantized rescale, faithful to reference fp32 math ----
  const float exp_scale = *exp_scale_p;
  const float exp_zero = *exp_zero_p;
  const float normed_scale = *normed_scale_p;
  const float normed_zero = *normed_zero_p;
  const float mult = exp_scale * (1.0f / exp_scale) / normed_scale;
  const float denom = 1.0f / rowsum;
  const float bz = denom - 1.0f / exp_zero;

  float4* __restrict__ outRow = (float4*)(out + base);
#pragma unroll
  for (int i = 0; i < ITER; ++i) {
    float4 o;
    o.x = fminf(fmaxf(__builtin_rintf(((e[4 * i + 0] - exp_zero) * bz) * mult + normed_zero), 0.0f), 255.0f);
    o.y = fminf(fmaxf(__builtin_rintf(((e[4 * i + 1] - exp_zero) * bz) * mult + normed_zero), 0.0f), 255.0f);
    o.z = fminf(fmaxf(__builtin_rintf(((e[4 * i + 2] - exp_zero) * bz) * mult + normed_zero), 0.0f), 255.0f);
    o.w = fminf(fmaxf(__builtin_rintf(((e[4 * i + 3] - exp_zero) * bz) * mult + normed_zero), 0.0f), 255.0f);
    outRow[i * 32 + lane] = o;
  }
}

extern "C" void kernel_launch(void* const* d_in, const int* in_sizes, int n_in,
                              void* d_out, int out_size, void* d_ws,
                              size_t ws_size, hipStream_t stream) {
  const int* inp = (const int*)d_in[0];
  const float* lut = (const float*)d_in[1];
  const float* exp_scale = (const float*)d_in[2];
  const float* exp_zero = (const float*)d_in[3];
  const float* normed_scale = (const float*)d_in[4];
  const float* normed_zero = (const float*)d_in[5];
  float* out = (float*)d_out;

  dim3 grid(NROWS / WAVES_PER_BLOCK);  // 2048 blocks x 256 threads (8 waves)
  qsoftmax_gfx1250<<<grid, 256, 0, stream>>>(inp, lut, exp_scale, exp_zero,
                                             normed_scale, normed_zero, out);
}